// SemanticFeedbackLoss_17875653886597
// MI455X (gfx1250) — compile-verified
//
#include <hip/hip_runtime.h>
#include <hip/hip_bf16.h>

typedef __attribute__((ext_vector_type(2))) float v2f;
typedef __attribute__((ext_vector_type(8))) float v8f;

#define DIMS 512
#define LDS_STRIDE 68   // padded row stride in floats: conflict-free & 16B-aligned
#define CHUNK 64        // K-chunk staged in LDS per iteration
#define LAMBDA_SEMANTIC 0.01f

// ---------------------------------------------------------------------------
// Kernel 1: per-row inverse norms. One wave (32 lanes) per 512-float row.
// ---------------------------------------------------------------------------
__global__ __launch_bounds__(256) void row_rnorm_kernel(
    const float* __restrict__ cb, float* __restrict__ rnorm, int V) {
  int wave = threadIdx.x >> 5;
  int lane = threadIdx.x & 31;
  int row  = blockIdx.x * 8 + wave;
  if (row >= V) return;
  const float4* p = (const float4*)(cb + (size_t)row * DIMS);
  float ss = 0.f;
#pragma unroll
  for (int j = 0; j < 4; ++j) {
    float4 v = p[j * 32 + lane];              // 512B contiguous per wave iter
    ss += v.x * v.x + v.y * v.y + v.z * v.z + v.w * v.w;
  }
#pragma unroll
  for (int off = 16; off >= 1; off >>= 1) ss += __shfl_xor(ss, off, 32);
  if (lane == 0) rnorm[row] = 1.0f / fmaxf(sqrtf(ss), 1e-12f);
}

// ---------------------------------------------------------------------------
// Kernel 2: 16 pairs per wave via V_WMMA_F32_16X16X4_F32; diagonal of the
// 16x16 accumulator holds the 16 pair dot products.
// ---------------------------------------------------------------------------
__global__ __launch_bounds__(256) void pair_wmma_kernel(
    const float* __restrict__ cb, const float* __restrict__ scores,
    const int* __restrict__ idx1, const int* __restrict__ idx2,
    const float* __restrict__ rnorm,
    float* __restrict__ psum, float* __restrict__ pcnt, int nGroups) {
  __shared__ float lds[8][2][16 * LDS_STRIDE];
  __shared__ float wsum[8], wcnt[8];
  int wave  = threadIdx.x >> 5;
  int lane  = threadIdx.x & 31;
  int group = blockIdx.x * 8 + wave;

  float sum = 0.f, cnt = 0.f;
  if (group < nGroups) {                      // wave-uniform: EXEC stays all-1s
    float* ldsA = lds[wave][0];
    float* ldsB = lds[wave][1];
    int p0 = group * 16;
    int myi1 = idx1[p0 + (lane & 15)];        // lane L caches pair (L&15) idx
    int myi2 = idx2[p0 + (lane & 15)];

    v8f acc = {};
    int halfrow = lane >> 4;                  // 0 or 1
    int col     = lane & 15;
    int rowf    = (lane & 15) * LDS_STRIDE;   // WMMA operand row base
    int ko      = (lane < 16) ? 0 : 2;        // K offset per half-wave

    for (int c = 0; c < 8; ++c) {
      int kbase = c * CHUNK;
      // ---- stage 16x64 A-rows and B-rows: each half-wave loads one 256B
      //      contiguous row segment as float4 (fully coalesced) -------------
#pragma unroll
      for (int r2 = 0; r2 < 8; ++r2) {
        int row = r2 * 2 + halfrow;
        int ia  = __shfl(myi1, row, 32);
        int ib  = __shfl(myi2, row, 32);
        float4 va = *(const float4*)(cb + (size_t)ia * DIMS + kbase + col * 4);
        float4 vb = *(const float4*)(cb + (size_t)ib * DIMS + kbase + col * 4);
        *(float4*)(&ldsA[row * LDS_STRIDE + col * 4]) = va;
        *(float4*)(&ldsB[row * LDS_STRIDE + col * 4]) = vb;
      }
      // ---- 16 WMMAs over this K-chunk; ds_load_b64 per operand,
      //      bank-conflict-free thanks to stride 68 ------------------------
#pragma unroll
      for (int k0 = 0; k0 < CHUNK; k0 += 4) {
        v2f a = *(const v2f*)(&ldsA[rowf + k0 + ko]);
        v2f b = *(const v2f*)(&ldsB[rowf + k0 + ko]);
        acc = __builtin_amdgcn_wmma_f32_16x16x4_f32(
            false, a, false, b, (short)0, acc, false, false);
      }
    }

    // ---- diagonal extraction ------------------------------------------------
    // C/D layout: VGPR g, lanes 0-15 -> (M=g, N=lane); lanes 16-31 -> (M=g+8,
    // N=lane-16).  Diagonal: lane m (m<8) holds acc[m]; lane 24+g holds acc[g]
    // for pair 8+g.
    int g = lane & 7;
    float dv = acc[0];
    dv = (g == 1) ? acc[1] : dv;
    dv = (g == 2) ? acc[2] : dv;
    dv = (g == 3) ? acc[3] : dv;
    dv = (g == 4) ? acc[4] : dv;
    dv = (g == 5) ? acc[5] : dv;
    dv = (g == 6) ? acc[6] : dv;
    dv = (g == 7) ? acc[7] : dv;
    bool isDiag = (lane < 8) || (lane >= 24);
    int p   = (lane < 8) ? lane : ((lane >= 24) ? (lane - 16) : 0);
    int i1p = __shfl(myi1, p, 32);
    int i2p = __shfl(myi2, p, 32);
    float sc = scores[p0 + p];
    float rn = rnorm[i1p] * rnorm[i2p];
    bool valid = isDiag && (i1p != i2p);
    sum = valid ? dv * sc * rn : 0.f;
    cnt = valid ? 1.f : 0.f;
#pragma unroll
    for (int off = 16; off >= 1; off >>= 1) {
      sum += __shfl_xor(sum, off, 32);
      cnt += __shfl_xor(cnt, off, 32);
    }
  }
  if (lane == 0) { wsum[wave] = sum; wcnt[wave] = cnt; }
  __syncthreads();
  if (threadIdx.x == 0) {
    float s = 0.f, cc = 0.f;
#pragma unroll
    for (int i = 0; i < 8; ++i) { s += wsum[i]; cc += wcnt[i]; }
    psum[blockIdx.x] = s;
    pcnt[blockIdx.x] = cc;
  }
}

// ---------------------------------------------------------------------------
// Kernel 3: reduce per-block partials -> scalar loss (deterministic, no
// float atomics anywhere).
// ---------------------------------------------------------------------------
__global__ __launch_bounds__(256) void finalize_kernel(
    const float* __restrict__ psum, const float* __restrict__ pcnt,
    float* __restrict__ out, int n) {
  __shared__ float ss[256], sc[256];
  float s = 0.f, c = 0.f;
  for (int i = threadIdx.x; i < n; i += 256) { s += psum[i]; c += pcnt[i]; }
  ss[threadIdx.x] = s; sc[threadIdx.x] = c;
  __syncthreads();
  for (int off = 128; off >= 1; off >>= 1) {
    if (threadIdx.x < off) {
      ss[threadIdx.x] += ss[threadIdx.x + off];
      sc[threadIdx.x] += sc[threadIdx.x + off];
    }
    __syncthreads();
  }
  if (threadIdx.x == 0) {
    float cv = sc[0];
    out[0] = (cv > 0.f) ? (-LAMBDA_SEMANTIC * ss[0] / cv) : 0.f;
  }
}

extern "C" void kernel_launch(void* const* d_in, const int* in_sizes, int n_in,
                              void* d_out, int out_size, void* d_ws, size_t ws_size,
                              hipStream_t stream) {
  const float* cb     = (const float*)d_in[0];
  const float* scores = (const float*)d_in[1];
  const int*   idx1   = (const int*)d_in[2];
  const int*   idx2   = (const int*)d_in[3];

  int V = in_sizes[0] / DIMS;       // 65536
  int N = in_sizes[1];              // 262144
  int nGroups    = N / 16;          // 16384 (N is a multiple of 16)
  int pairBlocks = (nGroups + 7) / 8;
  int normBlocks = (V + 7) / 8;

  float* rnorm = (float*)d_ws;            // V floats
  float* psum  = rnorm + V;               // pairBlocks floats
  float* pcnt  = psum + pairBlocks;       // pairBlocks floats

  row_rnorm_kernel<<<normBlocks, 256, 0, stream>>>(cb, rnorm, V);
  pair_wmma_kernel<<<pairBlocks, 256, 0, stream>>>(cb, scores, idx1, idx2,
                                                   rnorm, psum, pcnt, nGroups);
  finalize_kernel<<<1, 256, 0, stream>>>(psum, pcnt, (float*)d_out, pairBlocks);
}